// Heat_forward_21294447853933
// MI455X (gfx1250) — compile-verified
//
#include <hip/hip_runtime.h>

// ---------------------------------------------------------------------------
// Crank-Nicolson heat step, matrix-free on MI455X (gfx1250, wave32).
//
//   A0 = (1-2r)I + (r/2)S,  A1 = (1+2r)I - (r/2)S,   S = E(+-1) + E(+-64)
//   (flat-index bands: the +-1 band wraps across grid-row ends, exactly like
//   the dense np.diag construction in the reference).
//
//   b = A0 u + (dt/2)(f0+f1);  A1 x = b via Chebyshev-accelerated Jacobi
//   over [-rho, rho], rho = 2r/(1+2r) (Gershgorin, safe upper bound):
//     x1 = f,  x2 = w2(G x1 + f),  x_{k+1} = w(G x_k + f - x_{k-1}) + x_{k-1}
//   18 sweeps -> ~1e-11 residual, far below fp32 eps. Output boundary-masked.
//
// One workgroup (256 thr = 8 waves) per batch image; ~49 KB LDS-resident.
// Input planes arrive via three Tensor Data Mover descriptors (stride-3
// gather, global->LDS, waves 0-2, TENSORcnt). Vertical coupling (flat +-64)
// = tridiag(1,0,1) @ U on the matrix pipe (V_WMMA_F32_16X16X4_F32);
// horizontal (flat +-1) on the VALU; the pipes co-execute (XDL WMMA is
// TRANS-tracked). Zero-halo buffers keep stencil reads branch-free; one
// barrier per sweep (ping-pong buffers double as the x_{k-1} term).
// ---------------------------------------------------------------------------

#define NX     64
#define NODE   4096
#define NBATCH 256
#define NPAIRS 8        // Chebyshev sweeps = 2 + 2*NPAIRS = 18

#if defined(__AMDGCN__)
#  define USE_TDM 1     // tensor_load_to_lds inline-asm path (probe-verified)
#else
#  define USE_TDM 0
#endif

typedef __attribute__((ext_vector_type(2))) float        v2f;
typedef __attribute__((ext_vector_type(8))) float        v8f;
typedef __attribute__((ext_vector_type(4))) unsigned int u32x4;
typedef __attribute__((ext_vector_type(8))) unsigned int u32x8;

#if defined(__has_builtin)
#  if __has_builtin(__builtin_amdgcn_wmma_f32_16x16x4_f32)
#    define HAVE_WMMA_F32 1
#  endif
#endif
#ifndef HAVE_WMMA_F32
#  define HAVE_WMMA_F32 0
#endif

// Per-lane bitmasks for the T16 = tridiag(1,0,1) A-fragments, one bit per
// lane, components [slice s][vgpr c].  A 16x4 f32 layout (ISA 7.12.2):
// lanes 0-15 hold M=0..15 with {K0|K2} in VGPR0, {K1|K3} in VGPR1; bit set
// iff |M - K| == 1 for that lane's (M, K = 4s + 2*(lane>>4) + c).
__constant__ const unsigned int kAMask[4][2] = {
  {0x000A0002u, 0x00140005u},   // s=0: K={0|2}, {1|3}
  {0x00A00028u, 0x01400050u},   // s=1: K={4|6}, {5|7}
  {0x0A000280u, 0x14000500u},   // s=2: K={8|10}, {9|11}
  {0xA0002800u, 0x40005000u},   // s=3: K={12|14}, {13|15}
};

struct f3 { float x, y, z; };   // 12 B packed -> global_load_b96 (fallback)

__global__ __launch_bounds__(256) void heat_cn_kernel(const float* __restrict__ in,
                                                      float* __restrict__ out) {
  const float r    = 1e-4f * (float)(NX * NX);  // ALPHA*DT/dx^2 = 0.4096
  const float CR   = 0.5f * r;                  // r/2
  const float D0   = 1.0f - 2.0f * r;           // A0 diagonal
  const float C0   = 1.0f / (1.0f + 2.0f * r);  // 1 / A1 diagonal
  const float HDT  = 0.5f * 1e-4f;              // dt/2
  const float RHO  = 4.0f * CR * C0;            // Gershgorin bound, ~0.4503
  const float RHO2 = RHO * RHO;

  // zero-halo iterate buffers (64-elt halo each side); sB is owner-only
  __shared__ float sAh[NODE + 2 * NX];
  __shared__ float sBh[NODE + 2 * NX];
  __shared__ float sB [NODE];
  float* sXa = sAh + NX;   // u plane -> even Chebyshev iterates (final)
  float* sXb = sBh + NX;   // f1 plane -> odd Chebyshev iterates

  const int t    = threadIdx.x;
  const int lane = t & 31;
  const int wave = t >> 5;
  const int hi   = lane >> 4;
  const int n    = lane & 15;
  const int batch = blockIdx.x;

  // ---- zero halos (branchless stencil reads rely on these) ----
  if (t < 2 * NX) {
    const int h = (t < NX) ? (t - NX) : (NODE + t - NX);
    sXa[h] = 0.0f;
    sXb[h] = 0.0f;
  }

#if USE_TDM
  // ---- TDM: three stride-3 plane gathers, global -> LDS, waves 0..2 ----
  // D# per plane: data_size=4B, tile 1 x 4096, tensor_dim0_stride=3.
  if (wave < 3) {
    float* ldsPlane = (wave == 0) ? sXa : (wave == 1) ? sB : sXb;  // u, f0, f1
    const unsigned long long ga =
        (unsigned long long)(in + (size_t)batch * NODE * 3 + wave);
    const unsigned int lds = (unsigned int)(unsigned long long)ldsPlane;
    u32x4 g0;
    g0[0] = 1u;                                    // count=1, user descriptor
    g0[1] = lds;                                   // lds_addr (bytes)
    g0[2] = (unsigned int)ga;                      // global_addr[31:0]
    g0[3] = ((unsigned int)(ga >> 32) & 0x01FFFFFFu) | (2u << 30); // [56:32]|type=2
    u32x8 g1;
    g1[0] = 2u << 16;                              // data_size = 4 B
    g1[1] = 3u << 16;                              // tensor_dim0 = 3 (lo16)
    g1[2] = (unsigned)NODE << 16;                  // tensor_dim1 = 4096 (lo16)
    g1[3] = 1u << 16;                              // tile_dim0 = 1
    g1[4] = (unsigned)NODE;                        // tile_dim1 = 4096, tile_dim2 = 0
    g1[5] = 3u;                                    // tensor_dim0_stride = 3
    g1[6] = 0u;
    g1[7] = 0u;
    asm volatile("tensor_load_to_lds %0, %1" :: "s"(g0), "s"(g1) : "memory");
#if defined(__has_builtin) && __has_builtin(__builtin_amdgcn_s_wait_tensorcnt)
    __builtin_amdgcn_s_wait_tensorcnt(0);
#else
    asm volatile("s_wait_tensorcnt 0x0" ::: "memory");
#endif
  }
  __syncthreads();
#else
  // ---- fallback: coalesced b96 triplet loads, planar into LDS ----
  {
    const f3* __restrict__ src = (const f3*)(in + (size_t)batch * NODE * 3);
    if (t == 0) {
      __builtin_prefetch(src, 0, 3);
      __builtin_prefetch((const float*)src + 3 * 2048, 0, 3);
    }
#pragma unroll
    for (int k = 0; k < NODE / 256; ++k) {
      const int flat = t + 256 * k;
      const f3 v = src[flat];
      sXa[flat] = v.x;      // u
      sB [flat] = v.y;      // f0 (g is formed at RHS time)
      sXb[flat] = v.z;      // f1
    }
  }
  __syncthreads();
#endif

  // ---- A fragments from lane masks: 2-3 VALU each, cheap even if remat'd ----
  v2f afr[4];
#pragma unroll
  for (int s = 0; s < 4; ++s) {
    afr[s].x = ((kAMask[s][0] >> lane) & 1u) ? 1.0f : 0.0f;
    afr[s].y = ((kAMask[s][1] >> lane) & 1u) ? 1.0f : 0.0f;
  }

  // this wave owns two 16x16 tiles; S = vertical (WMMA) + horizontal (VALU)
  auto stencilTiles = [&](const float* __restrict__ src, float Sv[2][8]) {
#pragma unroll
    for (int ti = 0; ti < 2; ++ti) {
      const int tile = wave * 2 + ti;
      const int bi = tile >> 2, bj = tile & 3;
      const int colbase = 16 * bj + n;
#if HAVE_WMMA_F32
      v8f c = {};
#pragma unroll
      for (int s = 0; s < 4; ++s) {
        const int krow = 4 * s + hi * 2;  // B 4x16: VGPR0={row0|row2}, VGPR1={row1|row3}
        v2f bf;
        bf.x = src[(16 * bi + krow)     * NX + colbase];
        bf.y = src[(16 * bi + krow + 1) * NX + colbase];
        c = __builtin_amdgcn_wmma_f32_16x16x4_f32(false, afr[s], false, bf,
                                                  (short)0, c, false, false);
      }
#endif
#pragma unroll
      for (int vi = 0; vi < 8; ++vi) {
        const int m    = vi + 8 * hi;     // C/D 16x16: VGPRi = {M=i | M=i+8}
        const int flat = (16 * bi + m) * NX + colbase;
        float S;
#if HAVE_WMMA_F32
        S = c[vi];
        if (vi == 0) { const float up = src[flat - NX]; S += (hi == 0) ? up : 0.0f; }
        if (vi == 7) { const float dn = src[flat + NX]; S += (hi == 1) ? dn : 0.0f; }
#else
        S = src[flat - NX] + src[flat + NX];
#endif
        S += src[flat - 1] + src[flat + 1];   // flat +-1 band (halo-safe)
        Sv[ti][vi] = S;
      }
    }
  };

  float Sv[2][8];

  // ---- rhs + sweep 1: b = (1-2r)u + (r/2)Su + g ;  x1 = C0*b -> sXb ----
  stencilTiles(sXa, Sv);
#pragma unroll
  for (int ti = 0; ti < 2; ++ti) {
    const int tile = wave * 2 + ti;
    const int bi = tile >> 2, bj = tile & 3;
#pragma unroll
    for (int vi = 0; vi < 8; ++vi) {
      const int flat = (16 * bi + vi + 8 * hi) * NX + 16 * bj + n;
      const float g  = HDT * (sB[flat] + sXb[flat]);        // f0 + f1 planes
      const float b  = D0 * sXa[flat] + CR * Sv[ti][vi] + g;
      sB [flat] = b;           // each (ti,vi,lane) owns a unique element
      sXb[flat] = C0 * b;
    }
  }
  __syncthreads();

  // ---- sweep 2: x2 = w2*(G x1 + f) -> sXa (x0 = 0) ----
  float w = 1.0f / (1.0f - 0.5f * RHO2);
  stencilTiles(sXb, Sv);
#pragma unroll
  for (int ti = 0; ti < 2; ++ti) {
    const int tile = wave * 2 + ti;
    const int bi = tile >> 2, bj = tile & 3;
#pragma unroll
    for (int vi = 0; vi < 8; ++vi) {
      const int flat = (16 * bi + vi + 8 * hi) * NX + 16 * bj + n;
      sXa[flat] = w * (C0 * (sB[flat] + CR * Sv[ti][vi]));
    }
  }
  __syncthreads();

  // ---- Chebyshev pairs: x_{k+1} = w(G x_k + f - x_{k-1}) + x_{k-1} ----
  // (barrier-bounded sweeps: unrolling buys nothing, keep one copy)
#pragma unroll 1
  for (int p = 0; p < NPAIRS; ++p) {
    w = 1.0f / (1.0f - 0.25f * RHO2 * w);
    stencilTiles(sXa, Sv);
#pragma unroll
    for (int ti = 0; ti < 2; ++ti) {
      const int tile = wave * 2 + ti;
      const int bi = tile >> 2, bj = tile & 3;
#pragma unroll
      for (int vi = 0; vi < 8; ++vi) {
        const int flat = (16 * bi + vi + 8 * hi) * NX + 16 * bj + n;
        const float xp = sXb[flat];
        sXb[flat] = fmaf(w, C0 * (sB[flat] + CR * Sv[ti][vi]) - xp, xp);
      }
    }
    __syncthreads();

    w = 1.0f / (1.0f - 0.25f * RHO2 * w);
    stencilTiles(sXb, Sv);
#pragma unroll
    for (int ti = 0; ti < 2; ++ti) {
      const int tile = wave * 2 + ti;
      const int bi = tile >> 2, bj = tile & 3;
#pragma unroll
      for (int vi = 0; vi < 8; ++vi) {
        const int flat = (16 * bi + vi + 8 * hi) * NX + 16 * bj + n;
        const float xp = sXa[flat];
        sXa[flat] = fmaf(w, C0 * (sB[flat] + CR * Sv[ti][vi]) - xp, xp);
      }
    }
    __syncthreads();
  }

  // ---- Dirichlet mask + b128 store (final iterate lives in sXa) ----
  {
    float4* __restrict__ dst4 = (float4*)(out + (size_t)batch * NODE);
#pragma unroll
    for (int k = 0; k < NODE / (256 * 4); ++k) {
      const int q  = t + 256 * k;
      const int F  = q * 4;
      const int i  = F >> 6;
      const int j0 = F & (NX - 1);
      const bool rb = (i == 0) | (i == NX - 1);
      float4 v;
      v.x = (rb || j0 == 0)          ? 0.0f : sXa[F + 0];
      v.y = rb                       ? 0.0f : sXa[F + 1];
      v.z = rb                       ? 0.0f : sXa[F + 2];
      v.w = (rb || j0 + 3 == NX - 1) ? 0.0f : sXa[F + 3];
      dst4[q] = v;
    }
  }
}

extern "C" void kernel_launch(void* const* d_in, const int* in_sizes, int n_in,
                              void* d_out, int out_size, void* d_ws, size_t ws_size,
                              hipStream_t stream) {
  (void)in_sizes; (void)n_in; (void)d_ws; (void)ws_size; (void)out_size;
  const float* in = (const float*)d_in[0];
  // d_in[1] = A1, d_in[2] = A0 are dense encodings of fixed 5-band stencils;
  // the kernel applies them matrix-free (r = ALPHA*DT*NX*NX at compile time).
  float* out = (float*)d_out;
  heat_cn_kernel<<<NBATCH, 256, 0, stream>>>(in, out);
}